// FeatureFusionMultiHeadAttention_88665304858995
// MI455X (gfx1250) — compile-verified
//
#include <hip/hip_runtime.h>
#include <hip/hip_bf16.h>
#include <cstdint>

typedef float v2f __attribute__((ext_vector_type(2)));
typedef float v8f __attribute__((ext_vector_type(8)));

#define B_  16
#define LQ_ 1024
#define LK_ 1024
#define D_  512
#define H_  8
#define DK_ 64

__device__ __forceinline__ v8f wmma_f32(v2f a, v2f b, v8f c) {
  // V_WMMA_F32_16X16X4_F32 : D = A(16x4) * B(4x16) + C(16x16), all fp32
  return __builtin_amdgcn_wmma_f32_16x16x4_f32(false, a, false, b, (short)0, c,
                                               false, false);
}

// ---------------------------------------------------------------------------
// LayerNorm: one 512-wide row per block of 256 threads (8 waves).
// ---------------------------------------------------------------------------
__global__ __launch_bounds__(256) void ln_kernel(const float* __restrict__ x,
                                                 const float* __restrict__ g,
                                                 const float* __restrict__ bb,
                                                 float* __restrict__ y) {
  const int row = blockIdx.x;
  const int t = threadIdx.x;
  const float* xr = x + (size_t)row * D_;
  float a0 = xr[t];
  float a1 = xr[t + 256];
  __shared__ float sm[8];
  float s = a0 + a1;
#pragma unroll
  for (int o = 16; o > 0; o >>= 1) s += __shfl_xor(s, o, 32);
  if ((t & 31) == 0) sm[t >> 5] = s;
  __syncthreads();
  float tot = 0.f;
#pragma unroll
  for (int w = 0; w < 8; ++w) tot += sm[w];
  const float mean = tot * (1.0f / D_);
  __syncthreads();
  const float d0 = a0 - mean, d1 = a1 - mean;
  s = d0 * d0 + d1 * d1;
#pragma unroll
  for (int o = 16; o > 0; o >>= 1) s += __shfl_xor(s, o, 32);
  if ((t & 31) == 0) sm[t >> 5] = s;
  __syncthreads();
  tot = 0.f;
#pragma unroll
  for (int w = 0; w < 8; ++w) tot += sm[w];
  const float rstd = rsqrtf(tot * (1.0f / D_) + 1e-5f);
  float* yr = y + (size_t)row * D_;
  yr[t]       = d0 * rstd * g[t] + bb[t];
  yr[t + 256] = d1 * rstd * g[t + 256] + bb[t + 256];
}

// ---------------------------------------------------------------------------
// Fused projection + RoPE. Per block (128 thr = 4 waves): one (b,h,16-row)
// tile. Three f32 WMMA GEMMs over D=512 (x*W, pe_sin*Wp, pe_cos*Wp; sin/cos
// share the B operand), bias add, rotate-every-two via lane shfl_xor(1).
// Output layout (B,H,L,DK).
// ---------------------------------------------------------------------------
__global__ __launch_bounds__(128) void proj_rope_kernel(
    const float* __restrict__ xln, const float* __restrict__ pe,
    const float* __restrict__ W, const float* __restrict__ bias,
    const float* __restrict__ Wp, const float* __restrict__ biasp,
    float* __restrict__ outp, int L) {
  const int ltiles = L >> 4;
  int bid = blockIdx.x;
  const int lt = bid % ltiles; bid /= ltiles;
  const int h = bid % H_;
  const int b = bid / H_;
  const int lane = threadIdx.x & 31;
  const int wid = threadIdx.x >> 5;
  const int m = lane & 15;       // A row / C column-in-tile
  const int kh = lane >> 4;      // K half select (K+0..1 vs K+2..3)
  const int l = lt * 16 + m;
  const int n = wid * 16 + m;    // output column (0..63)

  const float* xrow = xln + ((size_t)b * L + l) * D_;
  const float* p0 = pe + ((size_t)b * L + l) * D_;                 // sin source
  const float* p1 = pe + (((size_t)B_ + b) * L + l) * D_;          // cos source
  const float* W0 = W + (size_t)h * D_ * DK_;
  const float* Wp0 = Wp + (size_t)h * D_ * DK_;

  v8f cq = {}; v8f cs = {}; v8f cc = {};
  for (int k = 0; k < D_; k += 4) {
    const int ka = k + 2 * kh;
    v2f ax = { xrow[ka], xrow[ka + 1] };
    v2f a0 = { p0[ka], p0[ka + 1] };
    v2f a1 = { p1[ka], p1[ka + 1] };
    v2f bw = { W0[(size_t)ka * DK_ + n],  W0[(size_t)(ka + 1) * DK_ + n] };
    v2f bp = { Wp0[(size_t)ka * DK_ + n], Wp0[(size_t)(ka + 1) * DK_ + n] };
    cq = wmma_f32(ax, bw, cq);
    cs = wmma_f32(a0, bp, cs);
    cc = wmma_f32(a1, bp, cc);
  }
  const float bq  = bias[h * DK_ + n];
  const float bp2 = biasp[h * DK_ + n];
  const float sgn = (lane & 1) ? 1.0f : -1.0f;   // even n: -q[n+1]; odd: +q[n-1]
  v8f res;
#pragma unroll
  for (int r = 0; r < 8; ++r) {
    const float q  = cq[r] + bq;
    const float si = cs[r] + bp2;
    const float co = cc[r] + bp2;
    const float partner = __shfl_xor(q, 1, 32);  // q at column n^1 (same row)
    res[r] = q * co + sgn * partner * si;
  }
  const size_t obase = (((size_t)b * H_ + h) * L + (size_t)lt * 16) * DK_;
#pragma unroll
  for (int r = 0; r < 8; ++r)
    outp[obase + (size_t)(r + 8 * kh) * DK_ + n] = res[r];
}

// ---------------------------------------------------------------------------
// v = sat_ln @ Wv + bv  :  rows = B*LK flattened, 16-row tile per block.
// ---------------------------------------------------------------------------
__global__ __launch_bounds__(128) void vproj_kernel(const float* __restrict__ xln,
                                                    const float* __restrict__ Wv,
                                                    const float* __restrict__ bv,
                                                    float* __restrict__ outp) {
  const int rt = blockIdx.x;
  const int lane = threadIdx.x & 31;
  const int wid = threadIdx.x >> 5;
  const int m = lane & 15;
  const int kh = lane >> 4;
  const int n = wid * 16 + m;
  const float* xrow = xln + ((size_t)rt * 16 + m) * D_;
  v8f c = {};
  for (int k = 0; k < D_; k += 4) {
    const int ka = k + 2 * kh;
    v2f a = { xrow[ka], xrow[ka + 1] };
    v2f bf = { Wv[(size_t)ka * DK_ + n], Wv[(size_t)(ka + 1) * DK_ + n] };
    c = wmma_f32(a, bf, c);
  }
  const float bias = bv[n];
  const size_t obase = (size_t)rt * 16 * DK_;
#pragma unroll
  for (int r = 0; r < 8; ++r)
    outp[obase + (size_t)(r + 8 * kh) * DK_ + n] = c[r] + bias;
}

// ---------------------------------------------------------------------------
// Attention: per block (b,h,16 q-rows). scores (WMMA) -> LDS, softmax,
// write attn (transposed layout) to global, then attn@v (WMMA, A from LDS).
// ---------------------------------------------------------------------------
__global__ __launch_bounds__(128) void attn_kernel(
    const float* __restrict__ qr, const float* __restrict__ kr,
    const float* __restrict__ vbuf, float* __restrict__ attn_out,
    float* __restrict__ headb) {
  __shared__ float sc[16 * LK_];   // 64 KB score strip
  int bid = blockIdx.x;
  const int qt = bid % (LQ_ / 16); bid /= (LQ_ / 16);
  const int h = bid % H_;
  const int b = bid / H_;
  const int t = threadIdx.x;
  const int lane = t & 31;
  const int wid = t >> 5;
  const int m = lane & 15;
  const int kh = lane >> 4;

  // per-row column rotation so 16-row strided A reads don't serialize on banks
  auto sidx = [](int row, int col) {
    return row * LK_ + ((col + (row << 2)) & (LK_ - 1));
  };

  // ---- Stage 1: scores = (q k^T) / sqrt(DK) ----
  const size_t qbase = (((size_t)b * H_ + h) * LQ_ + (size_t)qt * 16) * DK_;
  v2f afr[16];                               // full-DK q fragments (32 VGPRs)
#pragma unroll
  for (int i = 0; i < 16; ++i) {
    const int ka = 4 * i + 2 * kh;
    afr[i].x = qr[qbase + (size_t)m * DK_ + ka];
    afr[i].y = qr[qbase + (size_t)m * DK_ + ka + 1];
  }
  for (int j = 0; j < 16; ++j) {             // 16 N-tiles of 16 keys per wave
    const int cb = wid * 256 + j * 16;
    const float* krow = kr + (((size_t)b * H_ + h) * LK_ + cb + m) * DK_;
    v8f c = {};
#pragma unroll
    for (int i = 0; i < 16; ++i) {
      const int ka = 4 * i + 2 * kh;
      v2f bf = { krow[ka], krow[ka + 1] };
      c = wmma_f32(afr[i], bf, c);
    }
#pragma unroll
    for (int r = 0; r < 8; ++r)
      sc[sidx(r + 8 * kh, cb + m)] = c[r] * 0.125f;
  }
  __syncthreads();

  // ---- Stage 2: softmax per row (8 threads / row) ----
  const int row = t >> 3;
  const int sub = t & 7;
  float mx = -3.4e38f;
  for (int jj = 0; jj < 128; ++jj)
    mx = fmaxf(mx, sc[sidx(row, sub + 8 * jj)]);
#pragma unroll
  for (int o = 1; o < 8; o <<= 1) mx = fmaxf(mx, __shfl_xor(mx, o, 32));
  float sum = 0.f;
  for (int jj = 0; jj < 128; ++jj) {
    const int idx = sidx(row, sub + 8 * jj);
    const float e = __expf(sc[idx] - mx);
    sc[idx] = e;
    sum += e;
  }
#pragma unroll
  for (int o = 1; o < 8; o <<= 1) sum += __shfl_xor(sum, o, 32);
  const float rs = 1.0f / sum;
  for (int jj = 0; jj < 128; ++jj) sc[sidx(row, sub + 8 * jj)] *= rs;
  __syncthreads();

  // ---- Stage 3: coalesced write of attn into (B,LQ,H,LK) output ----
  float* aout = attn_out + (((size_t)b * LQ_ + (size_t)qt * 16) * H_ + h) * LK_;
  for (int e = t; e < 16 * LK_; e += 128) {
    const int rrow = e >> 10;
    const int col = e & (LK_ - 1);
    aout[(size_t)rrow * H_ * LK_ + col] = sc[sidx(rrow, col)];
  }

  // ---- Stage 4: head = attn @ v  (A from LDS, WMMA over K=1024) ----
  const int n = wid * 16 + m;
  v8f hc = {};
  for (int k = 0; k < LK_; k += 4) {
    const int ka = k + 2 * kh;
    v2f a = { sc[sidx(m, ka)], sc[sidx(m, ka + 1)] };
    const float* vr = vbuf + ((size_t)b * LK_ + ka) * DK_;
    v2f bf = { vr[n], vr[DK_ + n] };
    hc = wmma_f32(a, bf, hc);
  }
  float* hb = headb + (((size_t)b * H_ + h) * LQ_ + (size_t)qt * 16) * DK_;
#pragma unroll
  for (int r = 0; r < 8; ++r)
    hb[(size_t)(r + 8 * kh) * DK_ + n] = hc[r];
}

// ---------------------------------------------------------------------------
// Mean over heads: hmean(B,LQ,DK) = (1/H) sum_h head(B,H,LQ,DK)
// ---------------------------------------------------------------------------
__global__ __launch_bounds__(256) void mean_kernel(const float* __restrict__ headb,
                                                   float* __restrict__ hmean) {
  const size_t i = (size_t)blockIdx.x * 256 + threadIdx.x;
  const size_t per = (size_t)LQ_ * DK_;
  const size_t b = i / per;
  const size_t rem = i - b * per;
  float s = 0.f;
#pragma unroll
  for (int h = 0; h < H_; ++h) s += headb[(b * H_ + h) * per + rem];
  hmean[i] = s * (1.0f / H_);
}

// ---------------------------------------------------------------------------
// out = hmean(B*LQ x 64) @ Wh(64 x 512). Block: 16 rows x 64 cols.
// ---------------------------------------------------------------------------
__global__ __launch_bounds__(128) void out_gemm_kernel(
    const float* __restrict__ hmean, const float* __restrict__ Wh,
    float* __restrict__ outp) {
  int bid = blockIdx.x;
  const int nb = bid % (D_ / 64);
  const int rt = bid / (D_ / 64);
  const int lane = threadIdx.x & 31;
  const int wid = threadIdx.x >> 5;
  const int m = lane & 15;
  const int kh = lane >> 4;
  const int n = nb * 64 + wid * 16 + m;
  const float* ar = hmean + ((size_t)rt * 16 + m) * DK_;
  v8f c = {};
#pragma unroll
  for (int k = 0; k < DK_; k += 4) {
    const int ka = k + 2 * kh;
    v2f a = { ar[ka], ar[ka + 1] };
    v2f bf = { Wh[(size_t)ka * D_ + n], Wh[(size_t)(ka + 1) * D_ + n] };
    c = wmma_f32(a, bf, c);
  }
  const size_t obase = (size_t)rt * 16 * D_;
#pragma unroll
  for (int r = 0; r < 8; ++r)
    outp[obase + (size_t)(r + 8 * kh) * D_ + n] = c[r];
}

// ---------------------------------------------------------------------------
extern "C" void kernel_launch(void* const* d_in, const int* in_sizes, int n_in,
                              void* d_out, int out_size, void* d_ws, size_t ws_size,
                              hipStream_t stream) {
  const float* satellite = (const float*)d_in[0];
  const float* sat_pe    = (const float*)d_in[1];
  const float* timeser   = (const float*)d_in[2];
  const float* ts_pe     = (const float*)d_in[3];
  const float* ln_sat_g  = (const float*)d_in[4];
  const float* ln_sat_b  = (const float*)d_in[5];
  const float* ln_ts_g   = (const float*)d_in[6];
  const float* ln_ts_b   = (const float*)d_in[7];
  const float* Wv   = (const float*)d_in[8];
  const float* bv   = (const float*)d_in[9];
  const float* Wq   = (const float*)d_in[10];
  const float* bq   = (const float*)d_in[11];
  const float* Wk   = (const float*)d_in[12];
  const float* bk   = (const float*)d_in[13];
  const float* Wsat = (const float*)d_in[14];
  const float* bsat = (const float*)d_in[15];
  const float* Wts  = (const float*)d_in[16];
  const float* bts  = (const float*)d_in[17];
  const float* Wh   = (const float*)d_in[18];

  const size_t SZ_BLD  = (size_t)B_ * LK_ * D_;        // 8,388,608
  const size_t SZ_BHLK = (size_t)B_ * H_ * LQ_ * DK_;  // 8,388,608
  const size_t SZ_BLDK = (size_t)B_ * LK_ * DK_;       // 1,048,576
  float* ws     = (float*)d_ws;
  float* sat_ln = ws;
  float* ts_ln  = sat_ln + SZ_BLD;
  float* qr     = ts_ln + SZ_BLD;
  float* kr     = qr + SZ_BHLK;
  float* vbuf   = kr + SZ_BHLK;
  float* headb  = vbuf + SZ_BLDK;
  float* hmean  = headb + SZ_BHLK;

  float* out  = (float*)d_out;
  float* attn = out + (size_t)B_ * LQ_ * D_;

  ln_kernel<<<B_ * LK_, 256, 0, stream>>>(satellite, ln_sat_g, ln_sat_b, sat_ln);
  ln_kernel<<<B_ * LQ_, 256, 0, stream>>>(timeser, ln_ts_g, ln_ts_b, ts_ln);
  proj_rope_kernel<<<B_ * H_ * (LQ_ / 16), 128, 0, stream>>>(
      ts_ln, ts_pe, Wq, bq, Wts, bts, qr, LQ_);
  proj_rope_kernel<<<B_ * H_ * (LK_ / 16), 128, 0, stream>>>(
      sat_ln, sat_pe, Wk, bk, Wsat, bsat, kr, LK_);
  vproj_kernel<<<(B_ * LK_) / 16, 128, 0, stream>>>(sat_ln, Wv, bv, vbuf);
  attn_kernel<<<B_ * H_ * (LQ_ / 16), 128, 0, stream>>>(qr, kr, vbuf, attn, headb);
  mean_kernel<<<(B_ * LQ_ * DK_) / 256, 256, 0, stream>>>(headb, hmean);
  out_gemm_kernel<<<((B_ * LQ_) / 16) * (D_ / 64), 128, 0, stream>>>(hmean, Wh, out);
}